// Group_27118423506986
// MI455X (gfx1250) — compile-verified
//
#include <hip/hip_runtime.h>

// Problem dims (fixed by reference)
#define B_   32
#define N_   16384
#define NG   128     // NUM_GROUP
#define GS   32      // GROUP_SIZE
#define CHUNK 512    // points per KNN chunk held in LDS
#define NCHUNK (N_ / CHUNK)
#define CDW  (CHUNK * 3)     // dwords per chunk (AoS xyz)
#define D2STRIDE 516         // padded row stride for d2 tile (bank decorrelation)

typedef float v2f __attribute__((ext_vector_type(2)));
typedef float v8f __attribute__((ext_vector_type(8)));
typedef unsigned int u32x4 __attribute__((ext_vector_type(4)));
typedef int i32x4 __attribute__((ext_vector_type(4)));
typedef int i32x8 __attribute__((ext_vector_type(8)));

#define FINF __builtin_inff()

#if __has_builtin(__builtin_amdgcn_tensor_load_to_lds) && \
    __has_builtin(__builtin_amdgcn_s_wait_tensorcnt)
#define USE_TDM 1
#endif

// ---------------------------------------------------------------------------
// Kernel 1: Farthest Point Sampling. One block per batch, 1024 threads
// (32 wave32s). Each thread owns 16 points in registers. 128 serial
// iterations: distance update + block-wide argmax (wave shfl + LDS tree).
// ---------------------------------------------------------------------------
__global__ __launch_bounds__(1024) void fps_kernel(const float* __restrict__ pts,
                                                   float* __restrict__ wsCenters,
                                                   float* __restrict__ outCenter) {
  const int b = blockIdx.x;
  const int t = threadIdx.x;
  const float* base = pts + (size_t)b * N_ * 3;

  float x[16], y[16], z[16], dist[16];
#pragma unroll
  for (int k = 0; k < 16; ++k) {
    int j = t + (k << 10);
    x[k] = base[j * 3 + 0];
    y[k] = base[j * 3 + 1];
    z[k] = base[j * 3 + 2];
    dist[k] = 1e10f;  // matches reference init
  }

  __shared__ float cen[3];
  __shared__ float redv[32];
  __shared__ int   redi[32];
  __shared__ int   curLds;

  const int lane = t & 31;
  const int wave = t >> 5;

  int cur = 0;  // pointnet2 convention: start at index 0
  for (int s = 0; s < NG; ++s) {
    if (t == (cur & 1023)) {
      int k = cur >> 10;
      cen[0] = x[k]; cen[1] = y[k]; cen[2] = z[k];
      float* wc = wsCenters + ((size_t)b * NG + s) * 3;
      wc[0] = x[k]; wc[1] = y[k]; wc[2] = z[k];
      float* oc = outCenter + ((size_t)b * NG + s) * 3;
      oc[0] = x[k]; oc[1] = y[k]; oc[2] = z[k];
    }
    __syncthreads();
    const float cx = cen[0], cy = cen[1], cz = cen[2];

    float bv = -1.0f;
    int   bi = 0x7fffffff;
#pragma unroll
    for (int k = 0; k < 16; ++k) {
      float dx = x[k] - cx, dy = y[k] - cy, dz = z[k] - cz;
      float d = dx * dx + dy * dy + dz * dz;
      d = fminf(d, dist[k]);
      dist[k] = d;
      int j = t + (k << 10);
      if (d > bv || (d == bv && j < bi)) { bv = d; bi = j; }
    }
#pragma unroll
    for (int off = 16; off; off >>= 1) {
      float ov = __shfl_xor(bv, off, 32);
      int   oi = __shfl_xor(bi, off, 32);
      if (ov > bv || (ov == bv && oi < bi)) { bv = ov; bi = oi; }
    }
    if (lane == 0) { redv[wave] = bv; redi[wave] = bi; }
    __syncthreads();
    if (wave == 0) {
      bv = redv[lane]; bi = redi[lane];
#pragma unroll
      for (int off = 16; off; off >>= 1) {
        float ov = __shfl_xor(bv, off, 32);
        int   oi = __shfl_xor(bi, off, 32);
        if (ov > bv || (ov == bv && oi < bi)) { bv = ov; bi = oi; }
      }
      if (lane == 0) curLds = bi;
    }
    __syncthreads();
    cur = curLds;
  }
}

#ifdef USE_TDM
// Issue a TDM load of `ndw` contiguous dwords from global -> LDS (wave-uniform).
// D# per CDNA5 ISA: group0 {count=1, lds_addr, global_addr, type=2},
// group1 {data_size=4B, tensor_dim0=tile_dim0=ndw, tensor_dim1=tile_dim1=1}.
// This toolchain's builtin takes 6 args (extra int32x8 group, then cpol).
__device__ __forceinline__ void tdm_load_1d(const float* gsrc, unsigned lds_addr, int ndw) {
  unsigned long long ga = (unsigned long long)(uintptr_t)gsrc;
  u32x4 g0;
  g0[0] = 1u;                                               // count=1, user mode
  g0[1] = lds_addr;                                         // lds_addr (bytes)
  g0[2] = (unsigned)ga;                                     // global_addr[31:0]
  g0[3] = (unsigned)((ga >> 32) & 0x01FFFFFFull) | (2u << 30); // [56:32], type=2
  i32x8 g1;
  g1[0] = (int)(2u << 16);                                  // data_size code 2 = 4B
  g1[1] = (int)(((unsigned)ndw & 0xFFFFu) << 16);           // tensor_dim0[15:0]
  g1[2] = (int)((((unsigned)ndw >> 16) & 0xFFFFu) | (1u << 16)); // tdim0 hi | tensor_dim1=1
  g1[3] = (int)(((unsigned)ndw & 0xFFFFu) << 16);           // tile_dim0 = ndw
  g1[4] = 1;                                                // tile_dim1=1, tile_dim2=0
  g1[5] = ndw;                                              // tensor_dim0_stride[31:0]
  g1[6] = (int)(((unsigned)ndw & 0xFFFFu) << 16);           // tensor_dim1_stride[15:0]
  g1[7] = 0;
  i32x4 gz4 = {0, 0, 0, 0};
  i32x8 gz8 = {0, 0, 0, 0, 0, 0, 0, 0};
  __builtin_amdgcn_tensor_load_to_lds(g0, g1, gz4, gz4, gz8, 0);
}
#endif

// ---------------------------------------------------------------------------
// Kernel 2: KNN via V_WMMA_F32_16X16X4_F32, TDM double-buffered point
// streaming, top-32 selection, gather. Grid: (NG/16, B), 256 threads.
// ---------------------------------------------------------------------------
__global__ __launch_bounds__(256) void knn_kernel(const float* __restrict__ pts,
                                                  const float* __restrict__ wsCenters,
                                                  float* __restrict__ outNb) {
  const int b     = blockIdx.y;
  const int gbase = blockIdx.x * 16;
  const int t     = threadIdx.x;
  const int lane  = t & 31;
  const int wave  = t >> 5;

  extern __shared__ float smem[];
  float* stg0 = smem;                     // [CDW] AoS xyz staging, buffer 0
  float* stg1 = stg0 + CDW;               // [CDW] buffer 1
  float* psq  = stg1 + CDW;               // [CHUNK]
  float* d2c  = psq + CHUNK;              // [16][D2STRIDE]
  float* topv = d2c + 16 * D2STRIDE;      // [16][32]
  int*   topi = (int*)(topv + 16 * 32);   // [16][32]
  float* ntv  = (float*)(topi + 16 * 32); // [16][32]
  int*   nti  = (int*)(ntv + 16 * 32);    // [16][32]

  const float* gpts = pts + (size_t)b * N_ * 3;

  // A operand: 16 centers x K=4 (x,y,z,0); 32-bit A layout:
  // lanes 0-15 hold K=0,1 ; lanes 16-31 hold K=2,3
  const float* wc = wsCenters + ((size_t)b * NG + gbase) * 3;
  v2f A;
  if (lane < 16) { A.x = wc[lane * 3 + 0]; A.y = wc[lane * 3 + 1]; }
  else           { A.x = wc[(lane - 16) * 3 + 2]; A.y = 0.0f; }

  for (int i = t; i < 16 * 32; i += 256) { topv[i] = FINF; topi[i] = 0x7fffffff; }

#ifdef USE_TDM
  if (wave == 0) tdm_load_1d(gpts, (unsigned)(uintptr_t)(void*)stg0, CDW);
#endif

  for (int ci = 0; ci < NCHUNK; ++ci) {
    const int c0 = ci * CHUNK;
    float* cur = (ci & 1) ? stg1 : stg0;

#ifdef USE_TDM
    if (wave == 0) __builtin_amdgcn_s_wait_tensorcnt(0);  // cur buffer complete
    __syncthreads();
    if (wave == 0 && ci + 1 < NCHUNK) {   // stream next chunk during compute
      float* nxt = (ci & 1) ? stg0 : stg1;
      tdm_load_1d(gpts + (size_t)(c0 + CHUNK) * 3,
                  (unsigned)(uintptr_t)(void*)nxt, CDW);
    }
#else
    __syncthreads();
    for (int i = t; i < CDW; i += 256) cur[i] = gpts[(size_t)c0 * 3 + i];
#endif
    // L2 warm-up two chunks ahead (global_prefetch_b8), one cacheline/thread
    if (ci + 2 < NCHUNK && t < 48)
      __builtin_prefetch(gpts + (size_t)(c0 + 2 * CHUNK) * 3 + t * 32, 0, 1);

    // ---- |x|^2 pass ----
#ifndef USE_TDM
    __syncthreads();
#endif
    for (int i = t; i < CHUNK; i += 256) {
      float a0 = cur[3 * i + 0], a1 = cur[3 * i + 1], a2 = cur[3 * i + 2];
      psq[i] = a0 * a0 + a1 * a1 + a2 * a2;
    }
    __syncthreads();

    // ---- WMMA: 32 tiles of 16 centers x 16 points; 4 per wave, unrolled ----
    const int l15   = lane & 15;
    const int xzoff = (lane < 16) ? 0 : 2;   // lanes>=16 supply z into K=2
#pragma unroll
    for (int u = 0; u < 4; ++u) {
      const int n0 = (wave + u * 8) * 16;
      float bx = cur[3 * (n0 + l15) + xzoff];
      float by = cur[3 * (n0 + l15) + 1];
      v2f Bv;
      Bv.x = bx;
      Bv.y = (lane < 16) ? by : 0.0f;        // K=3 zero-pad (branchless cndmask)
      v8f Cv = {};
      Cv = __builtin_amdgcn_wmma_f32_16x16x4_f32(
          false, A, false, Bv, (short)0, Cv, false, false);
      const int col  = n0 + l15;
      const float s  = psq[col];
      const int mofs = (lane < 16) ? 0 : 8;
#pragma unroll
      for (int r = 0; r < 8; ++r) {
        // d2 = |x|^2 - 2*dot  (|c|^2 constant per row -> ranking-invariant)
        d2c[(r + mofs) * D2STRIDE + col] = s - 2.0f * Cv[r];
      }
    }
    __syncthreads();

    // ---- selection: wave w owns centers w and w+8 ----
    for (int mi = 0; mi < 2; ++mi) {
      const int m = wave + mi * 8;
      float* row = d2c + m * D2STRIDE;
      float* tv = topv + m * 32;  int* ti = topi + m * 32;
      float* nv = ntv  + m * 32;  int* ni = nti  + m * 32;
      for (int k = 0; k < GS; ++k) {
        float bvv = FINF; int bg = 0x7fffffff; int bsrc = CHUNK + lane;
        for (int q = 0; q < CHUNK / 32; ++q) {
          int j = lane + q * 32;
          float v = row[j];
          int g = c0 + j;
          if (v < bvv || (v == bvv && g < bg)) { bvv = v; bg = g; bsrc = j; }
        }
        { float v = tv[lane]; int g = ti[lane];
          if (v < bvv || (v == bvv && g < bg)) { bvv = v; bg = g; bsrc = CHUNK + lane; } }
#pragma unroll
        for (int off = 16; off; off >>= 1) {
          float ov = __shfl_xor(bvv, off, 32);
          int   og = __shfl_xor(bg, off, 32);
          int   os = __shfl_xor(bsrc, off, 32);
          if (ov < bvv || (ov == bvv && og < bg)) { bvv = ov; bg = og; bsrc = os; }
        }
        if (bsrc < CHUNK) { if (lane == (bsrc & 31)) row[bsrc] = FINF; }
        else              { if (lane == (bsrc - CHUNK)) tv[bsrc - CHUNK] = FINF; }
        if (lane == 0) { nv[k] = bvv; ni[k] = bg; }
      }
      tv[lane] = nv[lane];
      ti[lane] = ni[lane];
    }
    __syncthreads();
  }

  // ---- gather neighbors + subtract center ----
  for (int mi = 0; mi < 2; ++mi) {
    const int m = wave + mi * 8;
    const int g = gbase + m;
    const float* wc2 = wsCenters + ((size_t)b * NG + g) * 3;
    const float cx = wc2[0], cy = wc2[1], cz = wc2[2];
    const int gidx = topi[m * 32 + lane];
    const float* p = gpts + (size_t)gidx * 3;
    float* o = outNb + (((size_t)b * NG + g) * GS + lane) * 3;
    o[0] = p[0] - cx;
    o[1] = p[1] - cy;
    o[2] = p[2] - cz;
  }
}

// ---------------------------------------------------------------------------
extern "C" void kernel_launch(void* const* d_in, const int* in_sizes, int n_in,
                              void* d_out, int out_size, void* d_ws, size_t ws_size,
                              hipStream_t stream) {
  (void)in_sizes; (void)n_in; (void)out_size; (void)ws_size;
  const float* pts = (const float*)d_in[0];
  float* out       = (float*)d_out;
  float* outNb     = out;                                  // [32,128,32,3]
  float* outCenter = out + (size_t)B_ * NG * GS * 3;       // [32,128,3]
  float* wsCenters = (float*)d_ws;                         // [32,128,3] scratch

  fps_kernel<<<B_, 1024, 0, stream>>>(pts, wsCenters, outCenter);

  dim3 grid(NG / 16, B_);
  const size_t shmem =
      (size_t)(2 * CDW + CHUNK + 16 * D2STRIDE + 4 * 16 * 32) * sizeof(float);
  knn_kernel<<<grid, 256, shmem, stream>>>(pts, wsCenters, outNb);
}